// Codebook_29334626631957
// MI455X (gfx1250) — compile-verified
//
#include <hip/hip_runtime.h>

// CDNA5 / gfx1250 VQ codebook quantization, v2.
// xe GEMM via V_WMMA_F32_16X16X4_F32 with C oriented [code(M) x row(N)].
// Each wave owns TWO 16-row N-tiles so every A-frag (codebook tile) fetched
// from LDS feeds two WMMAs -> halves DS bandwidth per FLOP (LDS banks are the
// binding limit at 4 waves/WGP issuing f32 WMMA).  -0.5*||e||^2 is folded into
// the accumulator init, so the epilogue is pure cmp/cndmask that co-executes
// with the XDL pipe.

typedef __attribute__((ext_vector_type(2))) float v2f;
typedef __attribute__((ext_vector_type(8))) float v8f;

#define K_CODES     512
#define DIM         64
#define CB_STRIDE   66        // padded floats per codebook row in LDS (bank spread)
#define ROWS_PER_WG 256       // 8 waves x 32 rows
#define N_PER_B     1024      // h*w

__global__ __launch_bounds__(256) void vq_argmin_kernel(
    const float* __restrict__ x,        // (64, 64, 32, 32) = (b, d, n)
    const float* __restrict__ cb,       // (512, 64)
    float* __restrict__ out)            // (b, d, n)
{
    __shared__ float sh_cb[K_CODES * CB_STRIDE];   // ~132 KB
    __shared__ float sh_nhe2[K_CODES];             // -0.5 * ||e_k||^2
    __shared__ int   sh_idx[ROWS_PER_WG];

    const int tid = threadIdx.x;
    const int wg  = blockIdx.x;
    const int b   = wg >> 2;                       // 4 WGs per batch image
    const int n0  = (wg & 3) * ROWS_PER_WG;

    // ---- Stage codebook into LDS (row-major, padded stride) ----
    for (int i = tid; i < K_CODES * (DIM / 4); i += 256) {
        int row = i >> 4;          // / (DIM/4)
        int c4  = i & 15;
        float4 v = reinterpret_cast<const float4*>(cb)[row * (DIM / 4) + c4];
        float* dst = &sh_cb[row * CB_STRIDE + c4 * 4];
        dst[0] = v.x; dst[1] = v.y; dst[2] = v.z; dst[3] = v.w;
    }
    __syncthreads();

    // ---- Precompute negated half squared norms ----
    for (int k = tid; k < K_CODES; k += 256) {
        const float* r = &sh_cb[k * CB_STRIDE];
        float s = 0.f;
#pragma unroll
        for (int d = 0; d < DIM; ++d) s += r[d] * r[d];
        sh_nhe2[k] = -0.5f * s;
    }
    __syncthreads();

    // ---- Per-wave row assignment: two 16-row N-tiles per wave ----
    const int wave  = tid >> 5;
    const int lane  = tid & 31;
    const int lhalf = lane & 15;
    const int hi    = lane >> 4;                   // 0: lanes 0-15, 1: lanes 16-31
    const int nrowA = n0 + wave * 32 + lhalf;      // tile A row (N index)
    const int nrowB = nrowA + 16;                  // tile B row
    const float* xb = x + (size_t)b * DIM * N_PER_B;

    // B-matrix frags (4x16 K x N, 32-bit): v0 = K{4s, 4s+2}, v1 = K{4s+1, 4s+3}
    v2f bfragA[16], bfragB[16];
#pragma unroll
    for (int s = 0; s < 16; ++s) {
        int d0 = 4 * s + 2 * hi;
        const float* p0 = xb + (size_t)d0 * N_PER_B;
        const float* p1 = xb + (size_t)(d0 + 1) * N_PER_B;
        bfragA[s].x = p0[nrowA];
        bfragA[s].y = p1[nrowA];
        bfragB[s].x = p0[nrowB];
        bfragB[s].y = p1[nrowB];
    }

    float bestA  = -3.4e38f, bestB  = -3.4e38f;
    int   bestcA = 0,        bestcB = 0;

    // ---- 32 code tiles of 16 codes; 16 k-steps each; A-frag reused 2x ----
#pragma unroll 1
    for (int ct = 0; ct < K_CODES / 16; ++ct) {
        // C layout: VGPR r, lanes 0-15 -> code ct*16+r ; lanes 16-31 -> ct*16+8+r
        const float* nh = &sh_nhe2[ct * 16 + 8 * hi];   // 32B-aligned
        float4 h0 = *reinterpret_cast<const float4*>(nh);
        float4 h1 = *reinterpret_cast<const float4*>(nh + 4);
        v8f accA = {h0.x, h0.y, h0.z, h0.w, h1.x, h1.y, h1.z, h1.w};
        v8f accB = accA;

#pragma unroll
        for (int s = 0; s < 16; ++s) {
            // A-matrix frag (16x4 M x K, 32-bit): lane L<16 = (M=L, K0/K1),
            // lane 16+L = (M=L, K2/K3)  -> one ds_load_b64 per lane, used twice
            const float* ap = &sh_cb[(ct * 16 + lhalf) * CB_STRIDE + 4 * s + 2 * hi];
            v2f afrag;
            afrag.x = ap[0];
            afrag.y = ap[1];
            accA = __builtin_amdgcn_wmma_f32_16x16x4_f32(
                false, afrag, false, bfragA[s], (short)0, accA, false, false);
            accB = __builtin_amdgcn_wmma_f32_16x16x4_f32(
                false, afrag, false, bfragB[s], (short)0, accB, false, false);
        }

        // acc[r] already holds xe - 0.5*||e||^2 ; strict > keeps lowest index
#pragma unroll
        for (int r = 0; r < 8; ++r) {
            int code = ct * 16 + 8 * hi + r;
            float sA = accA[r];
            if (sA > bestA) { bestA = sA; bestcA = code; }
            float sB = accB[r];
            if (sB > bestB) { bestB = sB; bestcB = code; }
        }
    }

    // ---- Merge the two lane halves (tie-break: lower code index) ----
    {
        float o = __shfl_xor(bestA, 16, 32);
        int   c = __shfl_xor(bestcA, 16, 32);
        if (o > bestA || (o == bestA && c < bestcA)) { bestA = o; bestcA = c; }
        o = __shfl_xor(bestB, 16, 32);
        c = __shfl_xor(bestcB, 16, 32);
        if (o > bestB || (o == bestB && c < bestcB)) { bestB = o; bestcB = c; }
    }
    if (lane < 16) {
        sh_idx[wave * 32 + lhalf]      = bestcA;
        sh_idx[wave * 32 + 16 + lhalf] = bestcB;
    }
    __syncthreads();

    // ---- Gather winning codebook rows to (b, d, n) output, coalesced in n ----
    for (int i = tid; i < ROWS_PER_WG * DIM; i += 256) {
        int d   = i >> 8;      // / 256
        int row = i & 255;
        int code = sh_idx[row];
        out[((size_t)b * DIM + d) * N_PER_B + (n0 + row)] = sh_cb[code * CB_STRIDE + d];
    }
}

extern "C" void kernel_launch(void* const* d_in, const int* in_sizes, int n_in,
                              void* d_out, int out_size, void* d_ws, size_t ws_size,
                              hipStream_t stream) {
    const float* x   = (const float*)d_in[0];   // (64,64,32,32) f32
    const float* cb  = (const float*)d_in[1];   // (512,64) f32
    float*       out = (float*)d_out;

    dim3 grid(256);   // 64 b * 4 row-tiles of 256 rows
    dim3 block(256);  // 8 waves
    hipLaunchKernelGGL(vq_argmin_kernel, grid, block, 0, stream, x, cb, out);
}